// PSROIPool_9809705304725
// MI455X (gfx1250) — compile-verified
//
#include <hip/hip_runtime.h>
#include <stdint.h>

// PS-ROI average pooling for MI455X (gfx1250).
// Memory-bound gather/reduce: no matmul FLOPs, so no WMMA applies.
// CDNA5 feature path used: async global->LDS DMA (ASYNCcnt) with per-wave
// double buffering, global_prefetch for ROI rows, wave32 shuffle reduction.

#define POOLED   7
#define OUT_DIM  21
#define NT       1029      // OUT_DIM*7*7 == C == outputs per ROI; channel == t
#define H_DIM    64
#define W_DIM    64
#define HW       4096
#define C_DIM    1029
#define SCALE    0.0625f
#define NISSUE   6         // fixed async issues per stage
#define SLOTS    (NISSUE * 32)   // 192 LDS f32 slots per stage buffer
#define WAVES    8

__device__ __forceinline__ int iclamp(int v, int lo, int hi) {
  return v < lo ? lo : (v > hi ? hi : v);
}

__global__ __launch_bounds__(256)
void psroi_async_kernel(const float* __restrict__ feat,
                        const float* __restrict__ rois,
                        float* __restrict__ out, int R) {
  __shared__ float stage[WAVES][2][SLOTS];   // 12 KB of 320 KB LDS
  const int r = blockIdx.x;
  if (r >= R) return;
  const int lane = threadIdx.x & 31;
  const int wv   = threadIdx.x >> 5;

  const float* roi = rois + (size_t)r * 5;
  __builtin_prefetch(roi, 0, 0);             // global_prefetch_b8
  const int   b       = (int)roi[0];
  const float start_w = rintf(roi[1]) * SCALE;          // rintf == round-half-even
  const float start_h = rintf(roi[2]) * SCALE;
  const float end_w   = rintf(roi[3] + 1.0f) * SCALE;
  const float end_h   = rintf(roi[4] + 1.0f) * SCALE;
  const float bin_h   = fmaxf(end_h - start_h, 0.1f) * (1.0f / POOLED);
  const float bin_w   = fmaxf(end_w - start_w, 0.1f) * (1.0f / POOLED);

  const float*   batchBase = feat + (size_t)b * C_DIM * HW;
  const uint32_t ldsBase0  = (uint32_t)(uintptr_t)&stage[wv][0][0];
  const uint32_t ldsBase1  = (uint32_t)(uintptr_t)&stage[wv][1][0];

  // Bin rectangle for flattened output/channel index t.
  auto binParams = [&](int t, int& h0, int& rh, int& w0, int& bw, int& lg) {
    const int ph = (t / POOLED) % POOLED;
    const int pw = t % POOLED;
    h0     = iclamp((int)floorf((float)ph       * bin_h + start_h), 0, H_DIM);
    int h1 = iclamp((int)ceilf ((float)(ph + 1) * bin_h + start_h), 0, H_DIM);
    w0     = iclamp((int)floorf((float)pw       * bin_w + start_w), 0, W_DIM);
    int w1 = iclamp((int)ceilf ((float)(pw + 1) * bin_w + start_w), 0, W_DIM);
    rh = h1 - h0;
    bw = w1 - w0;
    const int b1 = bw > 1 ? bw - 1 : 0;                 // bw <= 11 -> lg <= 4
    lg = b1 ? (32 - __clz(b1)) : 0;                     // RW = 1<<lg, pow2 >= bw
  };

  // Issue exactly NISSUE async B32 copies (one padded rh x (1<<lg) rectangle).
  auto issue = [&](int t, uint32_t ldsBase) {
    int h0, rh, w0, bw, lg;
    binParams(t, h0, rh, w0, bw, lg);
    (void)rh; (void)bw;
    const uint64_t gbase = (uint64_t)(uintptr_t)(batchBase + (size_t)t * HW);
    const int      mask  = (1 << lg) - 1;
    #pragma unroll
    for (int j = 0; j < NISSUE; ++j) {
      const int s  = lane + 32 * j;
      const int hh = iclamp(h0 + (s >> lg),   0, H_DIM - 1);  // clamp: always in-bounds
      const int ww = iclamp(w0 + (s & mask),  0, W_DIM - 1);
      const uint64_t ga = gbase + (uint64_t)(uint32_t)((hh << 6) + ww) * 4u;
      const uint32_t la = ldsBase + (uint32_t)s * 4u;
      asm volatile("global_load_async_to_lds_b32 %0, %1, off"
                   :: "v"(la), "v"(ga) : "memory");
    }
  };

  // Reduce one staged rectangle, masking the pad slots, and store the average.
  auto consume = [&](int t, const float* buf) {
    int h0, rh, w0, bw, lg;
    binParams(t, h0, rh, w0, bw, lg);
    (void)h0; (void)w0;
    const int mask = (1 << lg) - 1;
    float acc = 0.0f;
    #pragma unroll
    for (int j = 0; j < NISSUE; ++j) {
      const int s  = lane + 32 * j;
      const int ho = s >> lg;
      const int wo = s & mask;
      const float v = buf[s];                            // ds_load from LDS
      acc += (ho < rh && wo < bw) ? v : 0.0f;
    }
    #pragma unroll
    for (int o = 16; o > 0; o >>= 1)                     // wave32 reduction
      acc += __shfl_xor(acc, o, 32);
    if (lane == 0) {
      const int area = rh * bw;
      out[(size_t)r * NT + t] = (area > 0) ? acc / (float)area : 0.0f;
    }
  };

  // Software pipeline: per-wave double-buffered async DMA stream.
  issue(wv, ldsBase0);
  int bufIdx = 0;
  for (int t = wv; t < NT; t += WAVES) {
    const int tn = t + WAVES;
    if (tn < NT) {
      issue(tn, bufIdx ? ldsBase0 : ldsBase1);
      // 6 newest (next stage) may remain outstanding; async loads complete
      // in order, so <=6 outstanding implies current stage's data is in LDS.
      asm volatile("s_wait_asynccnt 6" ::: "memory");
    } else {
      asm volatile("s_wait_asynccnt 0" ::: "memory");
    }
    consume(t, bufIdx ? &stage[wv][1][0] : &stage[wv][0][0]);
    bufIdx ^= 1;
  }
}

extern "C" void kernel_launch(void* const* d_in, const int* in_sizes, int n_in,
                              void* d_out, int out_size, void* d_ws, size_t ws_size,
                              hipStream_t stream) {
  const float* feat = (const float*)d_in[0];
  const float* rois = (const float*)d_in[1];
  float* out = (float*)d_out;
  const int R = in_sizes[1] / 5;          // 512 ROIs, 5 floats each
  psroi_async_kernel<<<dim3(R), dim3(256), 0, stream>>>(feat, rois, out, R);
  (void)n_in; (void)out_size; (void)d_ws; (void)ws_size;
}